// dis_convolution_52243982189251
// MI455X (gfx1250) — compile-verified
//
#include <hip/hip_runtime.h>
#include <hip/hip_bf16.h>

#define H0 128
#define W0 512
#define ROWS_PER_BLK 4

typedef float __attribute__((ext_vector_type(4))) f4_t;

// ---------------------------------------------------------------------------
// Kernel A: build the distortion LUT bn[i,j] in float64, matching numpy:
//   bn = clip(trunc(i - (cos(atan(DX*(j-OX)/F)) - 1) * (i - OY)), 0, H0-1)
// Packed as uint8 (values in [0,127]) -> 64 KB table in d_ws.
// ---------------------------------------------------------------------------
__global__ __launch_bounds__(256) void build_bn_kernel(unsigned char* __restrict__ bn)
{
    const int idx = blockIdx.x * 256 + threadIdx.x;   // 0 .. 65535
    const int i = idx >> 9;                           // row    0..127
    const int j = idx & (W0 - 1);                     // column 0..511

    const double DX = 0.00859375;
    const double F  = 8.0;
    const double OX = 5.5 / 2.0;   // 2.75
    const double OY = 7.7 / 2.0;   // 3.85

    double t = DX * ((double)j - OX) / F;
    double c = cos(atan(t)) - 1.0;
    double v = trunc((double)i - c * ((double)i - OY));
    int b = (int)v;
    b = b < 0 ? 0 : (b > (H0 - 1) ? (H0 - 1) : b);
    bn[idx] = (unsigned char)b;
}

// ---------------------------------------------------------------------------
// Kernel B: the gather. One block = ROWS_PER_BLK rows of one (b,c) slice.
// blockDim = 512: thread t -> r = t>>7 (row in block), jq = t&127 (float4 col).
// bn rows for the block are staged into LDS via gfx1250 async load-to-LDS.
// ---------------------------------------------------------------------------
__global__ __launch_bounds__(512) void gather_kernel(const float* __restrict__ x,
                                                     const unsigned char* __restrict__ bn,
                                                     float* __restrict__ out)
{
    __shared__ unsigned char lds_bn[ROWS_PER_BLK * W0];   // 2048 bytes

    const int t      = threadIdx.x;   // 0..511
    const int rowBlk = blockIdx.x;    // 0..31
    const int bc     = blockIdx.y;    // 0..BC-1

    // ---- stage 2048 B of bn into LDS: 512 lanes x 4 B ----
    const unsigned int* gsrc =
        (const unsigned int*)(bn + (size_t)rowBlk * (ROWS_PER_BLK * W0));
#if defined(__gfx1250__) && __has_builtin(__builtin_amdgcn_global_load_async_to_lds_b32)
    {
        typedef __attribute__((address_space(1))) int* gptr_t;
        typedef __attribute__((address_space(3))) char* lcptr_t;
        typedef __attribute__((address_space(3))) int*  lptr_t;
        gptr_t g1 = (gptr_t)(gsrc + t);
        lptr_t l3 = (lptr_t)((lcptr_t)lds_bn + 4 * t);
        __builtin_amdgcn_global_load_async_to_lds_b32(g1, l3, 0, 0);
#if __has_builtin(__builtin_amdgcn_s_wait_asynccnt)
        __builtin_amdgcn_s_wait_asynccnt(0);
#else
        asm volatile("s_wait_asynccnt 0" ::: "memory");
#endif
    }
#else
    ((unsigned int*)lds_bn)[t] = gsrc[t];
#endif
    __syncthreads();

    const int r  = t >> 7;          // 0..3
    const int jq = t & 127;         // float4 index within the row
    const int i  = rowBlk * ROWS_PER_BLK + r;
    const int j0 = jq * 4;

    const uchar4 b4 = ((const uchar4*)lds_bn)[r * (W0 / 4) + jq];

    const float* __restrict__ src = x + (size_t)bc * (H0 * W0);

    f4_t v;
    if (b4.x == b4.y && b4.y == b4.z && b4.z == b4.w) {
        // fast path (>97% of quads): row index uniform across the quad
        v = *(const f4_t*)(src + (int)b4.x * W0 + j0);
    } else {
        v.x = src[(int)b4.x * W0 + j0 + 0];
        v.y = src[(int)b4.y * W0 + j0 + 1];
        v.z = src[(int)b4.z * W0 + j0 + 2];
        v.w = src[(int)b4.w * W0 + j0 + 3];
    }

    // streaming (non-temporal) store: don't pollute L2 with write-once output
    f4_t* dst = (f4_t*)(out + (size_t)bc * (H0 * W0) + (size_t)i * W0 + j0);
    __builtin_nontemporal_store(v, dst);
}

// ---------------------------------------------------------------------------
extern "C" void kernel_launch(void* const* d_in, const int* in_sizes, int n_in,
                              void* d_out, int out_size, void* d_ws, size_t ws_size,
                              hipStream_t stream)
{
    const float* x   = (const float*)d_in[0];
    float*       out = (float*)d_out;
    unsigned char* bn = (unsigned char*)d_ws;   // 64 KB LUT in workspace

    // number of (b,c) slices: total elements / (H0*W0)
    const int BC = in_sizes[0] / (H0 * W0);     // 16*64 = 1024

    // Kernel A: rebuild LUT (deterministic, exact float64 match to numpy)
    hipLaunchKernelGGL(build_bn_kernel, dim3((H0 * W0) / 256), dim3(256), 0, stream, bn);

    // Kernel B: fused gather
    hipLaunchKernelGGL(gather_kernel, dim3(H0 / ROWS_PER_BLK, BC), dim3(512), 0, stream,
                       x, bn, out);
}